// GCN_89163521065197
// MI455X (gfx1250) — compile-verified
//
#include <hip/hip_runtime.h>

typedef __attribute__((ext_vector_type(2))) float v2f;
typedef __attribute__((ext_vector_type(8))) float v8f;

// ---------------------------------------------------------------------------
// Kernel 1: degree accumulation  deg[dst] += 1  (edge weights are 1)
// ---------------------------------------------------------------------------
__global__ __launch_bounds__(256) void degree_kernel(
    const long long* __restrict__ dst, float* __restrict__ deg, int E) {
  int e = blockIdx.x * blockDim.x + threadIdx.x;
  if (e < E) atomicAdd(&deg[(int)dst[e]], 1.0f);
}

// ---------------------------------------------------------------------------
// Kernel 2: dinv = rsqrt(1+deg), selfw = 1/(1+deg)
// ---------------------------------------------------------------------------
__global__ __launch_bounds__(256) void dinv_kernel(
    const float* __restrict__ deg, float* __restrict__ dinv,
    float* __restrict__ selfw, int N) {
  int i = blockIdx.x * blockDim.x + threadIdx.x;
  if (i < N) {
    float d = 1.0f + deg[i];
    dinv[i]  = rsqrtf(d);
    selfw[i] = 1.0f / d;
  }
}

// ---------------------------------------------------------------------------
// Kernel 3: h = x @ W via V_WMMA_F32_16X16X4_F32 (exact f32 path, K padded 3->4)
// One wave32 owns one 16-row tile of h; HID=32 -> two 16-wide N tiles.
// A layout (16x4 f32): lanes 0-15 hold K=0,1; lanes 16-31 hold K=2,3 (K=3 == 0).
// B layout (4x16 f32): VGPR0 = rows K0/K2, VGPR1 = rows K1/K3, N striped on lanes.
// D layout: VGPR r -> (M=r, N=lane) lanes 0-15, (M=r+8, N=lane-16) lanes 16-31.
// Stores: wave-uniform full-tile fast path (no per-row guards), single base
// address + immediate offsets so the compiler can clause the 16 stores.
// ---------------------------------------------------------------------------
__global__ __launch_bounds__(256) void linear_wmma_kernel(
    const float* __restrict__ x, const float* __restrict__ W,
    float* __restrict__ h, int n_tiles, int N) {
  int wave = (int)((blockIdx.x * blockDim.x + threadIdx.x) >> 5);
  int lane = threadIdx.x & 31;
  if (wave >= n_tiles) return;          // wave-uniform: EXEC stays all-ones
  const int m0 = wave * 16;
  const int m  = lane & 15;
  const bool hi = lane >= 16;

  // A operand: row m0+m of x, channels selected per lane half (clamped load)
  int gm = m0 + m;
  const float* xr = x + (size_t)(gm < N ? gm : N - 1) * 3;
  v2f a;
  a.x = hi ? xr[2] : xr[0];
  a.y = hi ? 0.0f  : xr[1];

  // B operands for the two 16-column tiles of W (W is [3][32] row-major)
  const int n = lane & 15;
  v2f b0, b1;
  b0.x = hi ? W[2 * 32 + n]      : W[0 * 32 + n];
  b0.y = hi ? 0.0f               : W[1 * 32 + n];
  b1.x = hi ? W[2 * 32 + 16 + n] : W[0 * 32 + 16 + n];
  b1.y = hi ? 0.0f               : W[1 * 32 + 16 + n];

  v8f c0 = {}, c1 = {};
  c0 = __builtin_amdgcn_wmma_f32_16x16x4_f32(false, a, false, b0, (short)0,
                                             c0, false, false);
  c1 = __builtin_amdgcn_wmma_f32_16x16x4_f32(false, a, false, b1, (short)0,
                                             c1, false, false);

  const int rbase = hi ? 8 : 0;
  // Per-lane base: element (m0 + rbase, n); c1 lives at column n+16.
  float* hp = h + ((size_t)(m0 + rbase)) * 32 + n;

  if (m0 + 16 <= N) {
    // Full tile: 16 unguarded stores off one base address.
#pragma unroll
    for (int r = 0; r < 8; ++r) {
      hp[r * 32]      = c0[r];
      hp[r * 32 + 16] = c1[r];
    }
  } else {
    // Tail tile (never taken for N % 16 == 0, kept for generality).
#pragma unroll
    for (int r = 0; r < 8; ++r) {
      if (m0 + rbase + r < N) {
        hp[r * 32]      = c0[r];
        hp[r * 32 + 16] = c1[r];
      }
    }
  }
}

// ---------------------------------------------------------------------------
// Kernel 4: edge gather/scale/scatter-add. 32 lanes per edge, one channel per
// lane -> coalesced 128B row reads of h[src] and non-returning f32 atomics
// into agg[dst]. Both h and agg fit in the 192MB L2, so this stays on-chip.
// ---------------------------------------------------------------------------
__global__ __launch_bounds__(256) void scatter_kernel(
    const long long* __restrict__ src, const long long* __restrict__ dst,
    const float* __restrict__ dinv, const float* __restrict__ h,
    float* __restrict__ agg, int E) {
  int e = (int)((blockIdx.x * blockDim.x + threadIdx.x) >> 5);
  int c = threadIdx.x & 31;
  if (e >= E) return;
  int s = (int)src[e];
  int d = (int)dst[e];
  float nrm = dinv[s] * dinv[d];
  atomicAdd(&agg[(size_t)d * 32 + c], h[(size_t)s * 32 + c] * nrm);
}

// ---------------------------------------------------------------------------
// Kernel 5: agg += h*selfw + b (in place), and accumulate per-channel
// sum / sum-of-squares via LDS block reduction -> 64 global accumulators.
// ---------------------------------------------------------------------------
__global__ __launch_bounds__(256) void self_bias_stats_kernel(
    const float* __restrict__ h, const float* __restrict__ selfw,
    const float* __restrict__ b, float* __restrict__ agg,
    float* __restrict__ stats /* [64]: sum[32], sumsq[32] */, int N) {
  __shared__ float ssum[32];
  __shared__ float ssq[32];
  int tid = threadIdx.x;
  if (tid < 32) { ssum[tid] = 0.0f; ssq[tid] = 0.0f; }
  __syncthreads();

  int c    = tid & 31;
  int node = blockIdx.x * (blockDim.x >> 5) + (tid >> 5);
  float v = 0.0f;
  if (node < N) {
    size_t idx = (size_t)node * 32 + c;
    v = agg[idx] + h[idx] * selfw[node] + b[c];
    agg[idx] = v;
  }
  atomicAdd(&ssum[c], v);
  atomicAdd(&ssq[c], v * v);
  __syncthreads();
  if (tid < 32) {
    atomicAdd(&stats[tid], ssum[tid]);
    atomicAdd(&stats[32 + tid], ssq[tid]);
  }
}

// ---------------------------------------------------------------------------
// Kernel 6: BatchNorm (batch stats, biased var) + PReLU, writes output.
// ---------------------------------------------------------------------------
__global__ __launch_bounds__(256) void bn_prelu_kernel(
    const float* __restrict__ agg, const float* __restrict__ stats,
    const float* __restrict__ gamma, const float* __restrict__ beta,
    const float* __restrict__ pw, float* __restrict__ out, int N) {
  long long i = (long long)blockIdx.x * blockDim.x + threadIdx.x;
  long long total = (long long)N * 32;
  if (i >= total) return;
  int c = (int)(i & 31);
  float invN = 1.0f / (float)N;
  float mean = stats[c] * invN;
  float var  = stats[32 + c] * invN - mean * mean;
  float isd  = rsqrtf(var + 1e-5f);
  float y = (agg[i] - mean) * isd * gamma[c] + beta[c];
  float p = pw[0];
  out[i] = (y >= 0.0f) ? y : p * y;
}

// ---------------------------------------------------------------------------
// Launch
// inputs: 0:x [N,3] f32, 1:edge_index [2,E] i64, 2:W [3,32] f32, 3:b [32],
//         4:bn_gamma [32], 5:bn_beta [32], 6:prelu_w [1]
// ---------------------------------------------------------------------------
extern "C" void kernel_launch(void* const* d_in, const int* in_sizes, int n_in,
                              void* d_out, int out_size, void* d_ws,
                              size_t ws_size, hipStream_t stream) {
  const float*     x     = (const float*)d_in[0];
  const long long* edge  = (const long long*)d_in[1];
  const float*     W     = (const float*)d_in[2];
  const float*     bias  = (const float*)d_in[3];
  const float*     gamma = (const float*)d_in[4];
  const float*     beta  = (const float*)d_in[5];
  const float*     pw    = (const float*)d_in[6];

  const int N = in_sizes[0] / 3;
  const int E = in_sizes[1] / 2;
  const long long* src = edge;
  const long long* dst = edge + E;

  // Workspace layout (floats): deg[N] | dinv[N] | selfw[N] | h[N*32] |
  //                            agg[N*32] | stats[64]
  float* ws    = (float*)d_ws;
  float* deg   = ws;
  float* dinv  = deg + N;
  float* selfw = dinv + N;
  float* h     = selfw + N;
  float* agg   = h + (size_t)N * 32;
  float* stats = agg + (size_t)N * 32;

  hipMemsetAsync(deg, 0, (size_t)N * sizeof(float), stream);
  hipMemsetAsync(agg, 0, (size_t)N * 32 * sizeof(float), stream);
  hipMemsetAsync(stats, 0, 64 * sizeof(float), stream);

  degree_kernel<<<(E + 255) / 256, 256, 0, stream>>>(dst, deg, E);
  dinv_kernel<<<(N + 255) / 256, 256, 0, stream>>>(deg, dinv, selfw, N);

  const int n_tiles = (N + 15) / 16;            // 6250 for N=100000
  linear_wmma_kernel<<<(n_tiles + 7) / 8, 256, 0, stream>>>(x, W, h, n_tiles, N);

  // 8 edges per 256-thread block (32 lanes/edge)
  scatter_kernel<<<(E + 7) / 8, 256, 0, stream>>>(src, dst, dinv, h, agg, E);

  self_bias_stats_kernel<<<(N + 7) / 8, 256, 0, stream>>>(h, selfw, bias, agg,
                                                          stats, N);

  long long total = (long long)N * 32;
  bn_prelu_kernel<<<(int)((total + 255) / 256), 256, 0, stream>>>(
      agg, stats, gamma, beta, pw, (float*)d_out, N);
}